// NPCGAT_1468878815355
// MI455X (gfx1250) — compile-verified
//
#include <hip/hip_runtime.h>
#include <hip/hip_bf16.h>

// ---------------------------------------------------------------------------
// 2-layer GAT forward for gfx1250 (MI455X).
//  - GEMMs via V_WMMA_F32_16X16X4_F32 (fp32 WMMA, wave32, 16x16 tile / wave)
//  - edge softmax/aggregation via fp32 global atomics (memory-bound part)
// ---------------------------------------------------------------------------

typedef __attribute__((ext_vector_type(2))) float v2f;
typedef __attribute__((ext_vector_type(8))) float v8f;

#define N_NODES_F 128     // IN_FEATS and layer-1 hidden width (4 heads * 32)
#define GAT_H 4
#define GAT_D1 32
#define GAT_C 47
#define GAT_HD2 (GAT_H * GAT_C)   // 188
#define NEG_SLOPE 0.2f
#define ENC_NEG_INF 0x007FFFFFu   // encf(-inf)

// order-preserving float <-> uint encoding for atomic max
__device__ __forceinline__ unsigned encf(float f) {
  unsigned u = __float_as_uint(f);
  return (u & 0x80000000u) ? ~u : (u | 0x80000000u);
}
__device__ __forceinline__ float decf(unsigned e) {
  return __uint_as_float((e & 0x80000000u) ? (e & 0x7FFFFFFFu) : ~e);
}

// ---------------------------------------------------------------------------
// fp32 WMMA GEMM: C[M x N] = A[M x K] * B[K x N], row-major, K multiple of 4.
// One wave computes one 16x16 tile via V_WMMA_F32_16X16X4_F32, K-loop step 4.
// ---------------------------------------------------------------------------
__global__ __launch_bounds__(128) void gemm_wmma_f32(
    const float* __restrict__ A, const float* __restrict__ B,
    float* __restrict__ C, int M, int N, int K, int tilesN) {
  const int wave = (blockIdx.x * blockDim.x + threadIdx.x) >> 5;
  const int lane = threadIdx.x & 31;
  const int tilesM = (M + 15) >> 4;
  if (wave >= tilesM * tilesN) return;           // wave-uniform exit

  const int tm = wave / tilesN;
  const int tn = wave % tilesN;
  const int half = lane >> 4;                    // 0: K pair {0,1}, 1: {2,3}
  const int lm = lane & 15;

  const int row = tm * 16 + lm;                  // A row owned by this lane
  const int col = tn * 16 + lm;                  // B/D column owned by this lane
  const bool rOK = row < M;
  const bool cOK = col < N;
  const int rowc = rOK ? row : 0;                // clamped (keeps EXEC full)
  const int colc = cOK ? col : 0;

  const float* ap = A + (long)rowc * K + half * 2;
  const float* bp = B + (long)(half * 2) * N + colc;

  v8f acc = {0.f, 0.f, 0.f, 0.f, 0.f, 0.f, 0.f, 0.f};

  for (int k0 = 0; k0 < K; k0 += 4) {
    // A 16x4: lane l -> A[l%16][k0 + 2*(l/16) + {0,1}]
    v2f a;
    a.x = ap[0];
    a.y = ap[1];
    if (!rOK) { a.x = 0.f; a.y = 0.f; }
    // B 4x16 (mirror): lane l -> B[k0 + 2*(l/16) + {0,1}][l%16]
    v2f b;
    float b0 = bp[0];
    float b1 = bp[N];
    if (!cOK) { b0 = 0.f; b1 = 0.f; }
    b.x = b0; b.y = b1;

    acc = __builtin_amdgcn_wmma_f32_16x16x4_f32(
        false, a, false, b, (short)0, acc, false, false);

    ap += 4;
    bp += (long)4 * N;
  }

  // D layout: VGPR r -> M = tm*16 + r + 8*half, N = col
#pragma unroll
  for (int r = 0; r < 8; ++r) {
    int rm = tm * 16 + r + half * 8;
    if (rm < M && cOK) C[(long)rm * N + col] = acc[r];
  }
}

// el/er: per (node, head) attention logits  el = <feat[n,h,:], al[h,:]>
__global__ void elr_kernel(const float* __restrict__ feat,
                           const float* __restrict__ al,
                           const float* __restrict__ ar,
                           float* __restrict__ el, float* __restrict__ er,
                           int Nn, int D) {
  int t = blockIdx.x * blockDim.x + threadIdx.x;
  if (t >= Nn * GAT_H) return;
  int n = t / GAT_H, h = t % GAT_H;
  const float* f = feat + (long)n * GAT_H * D + h * D;
  float sl = 0.f, sr = 0.f;
  for (int d = 0; d < D; ++d) {
    float x = f[d];
    sl += x * al[h * D + d];
    sr += x * ar[h * D + d];
  }
  el[t] = sl;
  er[t] = sr;
}

__global__ void fill_u32(unsigned* __restrict__ p, long n, unsigned v) {
  long i = (long)blockIdx.x * blockDim.x + threadIdx.x;
  if (i < n) p[i] = v;
}

// pass 1: e = leakyrelu(el[src]+er[dst]); store e; segment-max into menc[dst]
__global__ void edge_max_kernel(const int* __restrict__ src,
                                const int* __restrict__ dst,
                                const float* __restrict__ el,
                                const float* __restrict__ er,
                                float* __restrict__ ebuf,
                                unsigned* __restrict__ menc, int E) {
  int e = blockIdx.x * blockDim.x + threadIdx.x;
  if (e >= E) return;
  int u = src[e], v = dst[e];
#pragma unroll
  for (int h = 0; h < GAT_H; ++h) {
    float val = el[u * GAT_H + h] + er[v * GAT_H + h];
    val = val > 0.f ? val : NEG_SLOPE * val;
    ebuf[e * GAT_H + h] = val;
    atomicMax(&menc[v * GAT_H + h], encf(val));
  }
}

// pass 2: ex = exp(e - m[dst]); store ex in place; segment-sum into s[dst]
__global__ void edge_exp_kernel(const int* __restrict__ dst,
                                float* __restrict__ ebuf,
                                const unsigned* __restrict__ menc,
                                float* __restrict__ s, int E) {
  int e = blockIdx.x * blockDim.x + threadIdx.x;
  if (e >= E) return;
  int v = dst[e];
#pragma unroll
  for (int h = 0; h < GAT_H; ++h) {
    float m = decf(menc[v * GAT_H + h]);
    if (!(m >= -3.0e38f && m <= 3.0e38f)) m = 0.f;  // isfinite fix (ref)
    float ex = expf(ebuf[e * GAT_H + h] - m);
    ebuf[e * GAT_H + h] = ex;
    atomicAdd(&s[v * GAT_H + h], ex);
  }
}

// pass 3: alpha = ex / max(s[dst], 1e-9)  (in place)
__global__ void edge_alpha_kernel(const int* __restrict__ dst,
                                  float* __restrict__ ebuf,
                                  const float* __restrict__ s, int E) {
  int e = blockIdx.x * blockDim.x + threadIdx.x;
  if (e >= E) return;
  int v = dst[e];
#pragma unroll
  for (int h = 0; h < GAT_H; ++h)
    ebuf[e * GAT_H + h] /= fmaxf(s[v * GAT_H + h], 1e-9f);
}

// pass 4: one wave per edge; agg[dst] += alpha[h] * feat[src]
__global__ __launch_bounds__(256) void edge_agg_kernel(
    const int* __restrict__ src, const int* __restrict__ dst,
    const float* __restrict__ feat, const float* __restrict__ alpha,
    float* __restrict__ agg, int E, int HD, int D) {
  int warp = (int)((blockIdx.x * (long)blockDim.x + threadIdx.x) >> 5);
  int lane = threadIdx.x & 31;
  if (warp >= E) return;
  int u = src[warp], v = dst[warp];
  const float* frow = feat + (long)u * HD;
  float* arow = agg + (long)v * HD;
  __builtin_prefetch(frow + lane, 0, 1);  // global_prefetch_b8
  for (int idx = lane; idx < HD; idx += 32) {
    int h = idx / D;
    float a = alpha[warp * GAT_H + h];
    atomicAdd(&arow[idx], frow[idx] * a);
  }
}

// layer-1 epilogue fused with NPFeatureShuffle: out[n] = elu(agg[inv[n]] + b)
__global__ void shuffle_elu_kernel(const float* __restrict__ agg,
                                   const float* __restrict__ b,
                                   const int* __restrict__ inv,
                                   float* __restrict__ outf, int Nn, int HD) {
  long t = (long)blockIdx.x * blockDim.x + threadIdx.x;
  if (t >= (long)Nn * HD) return;
  int n = (int)(t / HD), idx = (int)(t % HD);
  float v = agg[(long)inv[n] * HD + idx] + b[idx];
  outf[t] = v > 0.f ? v : expf(v) - 1.f;
}

// final: out[n,c] = mean over heads of (agg2[n,h,c] + b2[h,c])
__global__ void final_mean_kernel(const float* __restrict__ agg,
                                  const float* __restrict__ b2,
                                  float* __restrict__ out, int Nn) {
  int t = blockIdx.x * blockDim.x + threadIdx.x;
  if (t >= Nn * GAT_C) return;
  int n = t / GAT_C, c = t % GAT_C;
  float s = 0.f;
#pragma unroll
  for (int h = 0; h < GAT_H; ++h)
    s += agg[(long)n * GAT_HD2 + h * GAT_C + c] + b2[h * GAT_C + c];
  out[t] = s * (1.0f / GAT_H);
}

// ---------------------------------------------------------------------------
static inline int cdiv(long a, long b) { return (int)((a + b - 1) / b); }

extern "C" void kernel_launch(void* const* d_in, const int* in_sizes, int n_in,
                              void* d_out, int out_size, void* d_ws, size_t ws_size,
                              hipStream_t stream) {
  const float* x   = (const float*)d_in[0];
  const int*   src = (const int*)d_in[1];
  const int*   dst = (const int*)d_in[2];
  const int*   inv = (const int*)d_in[3];
  const float* W1  = (const float*)d_in[4];
  const float* al1 = (const float*)d_in[5];
  const float* ar1 = (const float*)d_in[6];
  const float* b1  = (const float*)d_in[7];
  const float* W2  = (const float*)d_in[8];
  const float* al2 = (const float*)d_in[9];
  const float* ar2 = (const float*)d_in[10];
  const float* b2  = (const float*)d_in[11];
  float* out = (float*)d_out;

  const int Nn = in_sizes[0] / N_NODES_F;  // 100000
  const int E  = in_sizes[1];              // 1600000
  const int F  = N_NODES_F;                // 128 (= K for both GEMMs)

  // workspace carve-out (floats)
  float* ws = (float*)d_ws;
  size_t o = 0;
  float* feat1 = ws + o; o += (size_t)Nn * F;        // layer1 projected feats / hshuf
  float* agg1  = ws + o; o += (size_t)Nn * F;        // layer1 aggregation
  float* feat2 = ws + o; o += (size_t)Nn * GAT_HD2;  // layer2 projected feats
  float* agg2  = ws + o; o += (size_t)Nn * GAT_HD2;  // layer2 aggregation
  float* ebuf  = ws + o; o += (size_t)E * GAT_H;     // e -> ex -> alpha (in place)
  float* el    = ws + o; o += (size_t)Nn * GAT_H;
  float* er    = ws + o; o += (size_t)Nn * GAT_H;
  float* sbuf  = ws + o; o += (size_t)Nn * GAT_H;
  unsigned* menc = (unsigned*)(ws + o);              // + Nn*GAT_H

  const int NH  = Nn * GAT_H;
  const int B256 = 256;

  // ===================== layer 1 =====================
  {
    const int tilesN = F / 16;                              // 8
    const long tiles = (long)((Nn + 15) / 16) * tilesN;
    gemm_wmma_f32<<<cdiv(tiles * 32, 128), 128, 0, stream>>>(
        x, W1, feat1, Nn, F, F, tilesN);
  }
  elr_kernel<<<cdiv(NH, B256), B256, 0, stream>>>(feat1, al1, ar1, el, er, Nn, GAT_D1);
  fill_u32<<<cdiv(NH, B256), B256, 0, stream>>>(menc, NH, ENC_NEG_INF);
  fill_u32<<<cdiv(NH, B256), B256, 0, stream>>>((unsigned*)sbuf, NH, 0u);
  fill_u32<<<cdiv((long)Nn * F, B256), B256, 0, stream>>>((unsigned*)agg1, (long)Nn * F, 0u);
  edge_max_kernel<<<cdiv(E, B256), B256, 0, stream>>>(src, dst, el, er, ebuf, menc, E);
  edge_exp_kernel<<<cdiv(E, B256), B256, 0, stream>>>(dst, ebuf, menc, sbuf, E);
  edge_alpha_kernel<<<cdiv(E, B256), B256, 0, stream>>>(dst, ebuf, sbuf, E);
  edge_agg_kernel<<<cdiv((long)E * 32, B256), B256, 0, stream>>>(
      src, dst, feat1, ebuf, agg1, E, F, GAT_D1);
  // elu(agg1 + b1) with cross-partition reindex -> overwrite feat1 (safe: agg done)
  shuffle_elu_kernel<<<cdiv((long)Nn * F, B256), B256, 0, stream>>>(
      agg1, b1, inv, feat1, Nn, F);

  // ===================== layer 2 =====================
  {
    const int tilesN = (GAT_HD2 + 15) / 16;                 // 12 (N=188 padded)
    const long tiles = (long)((Nn + 15) / 16) * tilesN;
    gemm_wmma_f32<<<cdiv(tiles * 32, 128), 128, 0, stream>>>(
        feat1, W2, feat2, Nn, GAT_HD2, F, tilesN);
  }
  elr_kernel<<<cdiv(NH, B256), B256, 0, stream>>>(feat2, al2, ar2, el, er, Nn, GAT_C);
  fill_u32<<<cdiv(NH, B256), B256, 0, stream>>>(menc, NH, ENC_NEG_INF);
  fill_u32<<<cdiv(NH, B256), B256, 0, stream>>>((unsigned*)sbuf, NH, 0u);
  fill_u32<<<cdiv((long)Nn * GAT_HD2, B256), B256, 0, stream>>>(
      (unsigned*)agg2, (long)Nn * GAT_HD2, 0u);
  edge_max_kernel<<<cdiv(E, B256), B256, 0, stream>>>(src, dst, el, er, ebuf, menc, E);
  edge_exp_kernel<<<cdiv(E, B256), B256, 0, stream>>>(dst, ebuf, menc, sbuf, E);
  edge_alpha_kernel<<<cdiv(E, B256), B256, 0, stream>>>(dst, ebuf, sbuf, E);
  edge_agg_kernel<<<cdiv((long)E * 32, B256), B256, 0, stream>>>(
      src, dst, feat2, ebuf, agg2, E, GAT_HD2, GAT_C);
  final_mean_kernel<<<cdiv(Nn * GAT_C, B256), B256, 0, stream>>>(agg2, b2, out, Nn);
}